// GCN_89635967467582
// MI455X (gfx1250) — compile-verified
//
#include <hip/hip_runtime.h>
#include <cstddef>

#define NNODES 100000
#define MTILES (NNODES / 16)   // 6250, exact

typedef __attribute__((ext_vector_type(2))) float v2f;
typedef __attribute__((ext_vector_type(8))) float v8f;

// ---------------- utility kernels ----------------

__global__ void fill_zero(float* __restrict__ p, int n) {
    int i = blockIdx.x * blockDim.x + threadIdx.x;
    if (i < n) p[i] = 0.0f;
}

__global__ void deg_accum(const int* __restrict__ dst, const float* __restrict__ w,
                          float* __restrict__ deg, int E) {
    int e = blockIdx.x * blockDim.x + threadIdx.x;
    if (e < E) unsafeAtomicAdd(&deg[dst[e]], w[e]);   // hw global_atomic_add_f32
}

__global__ void dinv_kernel(const float* __restrict__ deg, float* __restrict__ dinv, int n) {
    int i = blockIdx.x * blockDim.x + threadIdx.x;
    if (i < n) {
        float d = deg[i] + 1.0f;                      // +1 self loop
        dinv[i] = (d > 0.0f) ? rsqrtf(d) : 0.0f;
    }
}

// ---------------- fp32 WMMA GEMM: C[M,N] = A[M,K] @ W[K,N] (+bias) ----------------
// One wave computes one 16x16 output tile via V_WMMA_F32_16X16X4_F32.
// A 16x4 f32 layout: lanes 0-15 hold M=0..15 with K=k,k+1; lanes 16-31 same M with K=k+2,k+3.
// B 4x16 f32 layout mirrors A (rows striped across lanes within a VGPR).
// C/D: VGPR r -> M=r (lanes 0-15) / M=r+8 (lanes 16-31), N = lane&15.
template <int K, int N>
__global__ __launch_bounds__(256) void gemm_wmma(const float* __restrict__ A,
                                                 const float* __restrict__ W,
                                                 const float* __restrict__ bias,
                                                 float* __restrict__ C,
                                                 int add_bias) {
    const int lane  = threadIdx.x & 31;
    const int wave  = threadIdx.x >> 5;
    const int mtile = blockIdx.x * 8 + wave;
    if (mtile >= MTILES) return;                       // wave-uniform; EXEC stays all-1
    const int ntile = blockIdx.y;
    const int mr    = lane & 15;                        // A-row in tile / B-&C-column
    const int half  = lane >> 4;

    const float* arow = A + (size_t)(mtile * 16 + mr) * K;
    const float* wcol = W + ntile * 16 + mr;

    v8f acc = {};
#pragma unroll
    for (int k = 0; k < K; k += 4) {
        // 8-byte aligned, read-once -> nontemporal (don't evict edge list from L2)
        v2f a = __builtin_nontemporal_load((const v2f*)(arow + k + 2 * half));
        v2f b;
        b.x = wcol[(size_t)(k + 2 * half) * N];
        b.y = wcol[(size_t)(k + 2 * half + 1) * N];
        acc = __builtin_amdgcn_wmma_f32_16x16x4_f32(
            /*neg_a=*/false, a, /*neg_b=*/false, b,
            /*c_mod=*/(short)0, acc, /*reuse_a=*/false, /*reuse_b=*/false);
    }

    float bv = add_bias ? bias[ntile * 16 + mr] : 0.0f;
#pragma unroll
    for (int r = 0; r < 8; ++r) {
        int m = r + 8 * half;
        C[(size_t)(mtile * 16 + m) * N + ntile * 16 + mr] = acc[r] + bv;
    }
}

// ---------------- edge scatter: agg[dst] += dinv[src]*w*dinv[dst] * xw[src] ----------------
template <int H>
__global__ __launch_bounds__(256) void edge_scatter(const int* __restrict__ src,
                                                    const int* __restrict__ dst,
                                                    const float* __restrict__ w,
                                                    const float* __restrict__ dinv,
                                                    const float* __restrict__ xw,
                                                    float* __restrict__ agg, int E) {
    int e = blockIdx.x * blockDim.x + threadIdx.x;
    if (e >= E) return;
    int s = src[e], d = dst[e];
    float nrm = dinv[s] * w[e] * dinv[d];
    const float4* xs = (const float4*)(xw + (size_t)s * H);
    float* ad = agg + (size_t)d * H;
#pragma unroll
    for (int q = 0; q < H / 4; ++q) {
        float4 v = xs[q];                              // gather, L2-resident
        unsafeAtomicAdd(ad + q * 4 + 0, nrm * v.x);    // hw f32 atomics, L2-resident
        unsafeAtomicAdd(ad + q * 4 + 1, nrm * v.y);
        unsafeAtomicAdd(ad + q * 4 + 2, nrm * v.z);
        unsafeAtomicAdd(ad + q * 4 + 3, nrm * v.w);
    }
}

// ---------------- combine: out = ELU(agg + dinv^2*xw + b) ----------------
template <int H>
__global__ void combine_elu(const float* __restrict__ agg, const float* __restrict__ xw,
                            const float* __restrict__ dinv, const float* __restrict__ bias,
                            float* __restrict__ out, int total) {
    int idx = blockIdx.x * blockDim.x + threadIdx.x;
    if (idx >= total) return;
    int i = idx / H;
    int f = idx - i * H;
    float di = dinv[i];
    float v  = agg[idx] + di * di * xw[idx] + bias[f];
    out[idx] = (v > 0.0f) ? v : expm1f(v);
}

// ---------------- host launch ----------------
extern "C" void kernel_launch(void* const* d_in, const int* in_sizes, int n_in,
                              void* d_out, int out_size, void* d_ws, size_t ws_size,
                              hipStream_t stream) {
    const float* x  = (const float*)d_in[0];
    const int*   ei = (const int*)d_in[1];
    const float* ew = (const float*)d_in[2];
    const float* W0 = (const float*)d_in[3];
    const float* b0 = (const float*)d_in[4];
    const float* W1 = (const float*)d_in[5];
    const float* b1 = (const float*)d_in[6];
    const float* W2 = (const float*)d_in[7];
    const float* b2 = (const float*)d_in[8];
    const float* Wm = (const float*)d_in[9];
    const float* bm = (const float*)d_in[10];

    const int E = in_sizes[1] / 2;
    const int* src = ei;
    const int* dst = ei + E;

    char* ws = (char*)d_ws;
    float* deg  = (float*)(ws);                              // 0.4 MB
    float* dinv = (float*)(ws + (size_t)(1 << 19));          // 0.4 MB @ 0.5 MB
    float* xw   = (float*)(ws + (size_t)1  * (1 << 20));     // 12.8 MB @ 1 MB
    float* agg  = (float*)(ws + (size_t)14 * (1 << 20));     // 12.8 MB @ 14 MB
    float* hA   = (float*)(ws + (size_t)27 * (1 << 20));     // 12.8 MB @ 27 MB
    float* hB   = (float*)(ws + (size_t)40 * (1 << 20));     // 12.8 MB @ 40 MB

    dim3 blk(256);
    const int nbN   = (NNODES + 255) / 256;
    const int nbE   = (E + 255) / 256;
    const int nb16  = (NNODES * 16 + 255) / 256;
    const int nb32  = (NNODES * 32 + 255) / 256;
    const int gemmx = (MTILES + 7) / 8;                      // 782 blocks of 8 waves

    // degrees + dinv (edge weights fixed across layers -> compute once)
    fill_zero<<<nbN, blk, 0, stream>>>(deg, NNODES);
    deg_accum<<<nbE, blk, 0, stream>>>(dst, ew, deg, E);
    dinv_kernel<<<nbN, blk, 0, stream>>>(deg, dinv, NNODES);

    // ---- layer 0: [100000,256] @ [256,16] ----
    gemm_wmma<256, 16><<<dim3(gemmx, 1), blk, 0, stream>>>(x, W0, (const float*)nullptr, xw, 0);
    fill_zero<<<nb16, blk, 0, stream>>>(agg, NNODES * 16);
    edge_scatter<16><<<nbE, blk, 0, stream>>>(src, dst, ew, dinv, xw, agg, E);
    combine_elu<16><<<nb16, blk, 0, stream>>>(agg, xw, dinv, b0, hA, NNODES * 16);

    // ---- layer 1: [100000,16] @ [16,32] ----
    gemm_wmma<16, 32><<<dim3(gemmx, 2), blk, 0, stream>>>(hA, W1, (const float*)nullptr, xw, 0);
    fill_zero<<<nb32, blk, 0, stream>>>(agg, NNODES * 32);
    edge_scatter<32><<<nbE, blk, 0, stream>>>(src, dst, ew, dinv, xw, agg, E);
    combine_elu<32><<<nb32, blk, 0, stream>>>(agg, xw, dinv, b1, hB, NNODES * 32);

    // ---- layer 2: [100000,32] @ [32,32] ----
    gemm_wmma<32, 32><<<dim3(gemmx, 2), blk, 0, stream>>>(hB, W2, (const float*)nullptr, xw, 0);
    fill_zero<<<nb32, blk, 0, stream>>>(agg, NNODES * 32);
    edge_scatter<32><<<nbE, blk, 0, stream>>>(src, dst, ew, dinv, xw, agg, E);
    combine_elu<32><<<nb32, blk, 0, stream>>>(agg, xw, dinv, b2, hA, NNODES * 32);

    // ---- head: [100000,32] @ [32,16] + bm -> d_out ----
    gemm_wmma<32, 16><<<dim3(gemmx, 1), blk, 0, stream>>>(hA, Wm, bm, (float*)d_out, 1);
}